// HistogramLoss_13477607375612
// MI455X (gfx1250) — compile-verified
//
#include <hip/hip_runtime.h>

// HistogramLoss for MI455X (gfx1250, wave32).
//   Stage 1: zero workspace (hist_pos, hist_neg, class counts)
//   Stage 2: class-count histogram (analytic n_pos, avoids per-pair count atomics)
//   Stage 3: blocked Gram-matrix via V_WMMA_F32_16X16X4_F32 + in-register soft
//            binning into per-wave private LDS histograms, merged with one
//            global f32 atomic per block per bin
//   Stage 4: 151-bin prefix scan + dot -> scalar loss
//
// Working set (8.4 MB) lives in L2 (192 MB); the problem is compute bound, so
// we keep fp32 precision on the matrix pipe (bin width 0.0133 > bf16 error).
// Binning is division-free: 1/STEP == 75.0 exactly, and
// round((floor(s/STEP)*STEP + 1)/STEP) == floor(s*75) + 75 for s in [-1,1].

typedef float v2f __attribute__((ext_vector_type(2)));
typedef float v8f __attribute__((ext_vector_type(8)));

#define NN 8192
#define DD 256
#define BM 128         // rows per block
#define BN 64          // cols per block
#define BINS 151       // np.arange(-1, 1+2/150, 2/150) -> exactly 151 bins
#define BINS_PAD 152
#define NCLS 128       // classes are in [0, 100)

// ---------------------------------------------------------------- stage 1
__global__ void zero_ws_kernel(unsigned* ws) {
    int t = threadIdx.x;
    if (t < 2 * BINS_PAD + NCLS) ws[t] = 0u;
}

// ---------------------------------------------------------------- stage 2
__global__ void class_count_kernel(const int* __restrict__ cls, unsigned* cnt) {
    int i = blockIdx.x * blockDim.x + threadIdx.x;
    if (i < NN) atomicAdd(&cnt[cls[i] & (NCLS - 1)], 1u);
}

// ---------------------------------------------------------------- stage 3
// Soft-histogram one 16x16 f32 WMMA accumulator tile.
// C/D layout: lanes 0-15: VGPR v <-> M=v, N=lane; lanes 16-31: M=v+8, N=lane-16.
__device__ __forceinline__ void bin_tile(
    const v8f& c, int ibase, int jbase, int lane,
    const int* clsI, const int* clsJ, int row0, int col0,
    float* hp, float* hn)
{
    const float STEP = 2.0f / 150.0f;
    const float INV  = 75.0f;          // 1/STEP, exact in f32
    const int hi = lane >> 4;
    const int n  = lane & 15;
    const int j  = jbase + n;
    const int cj = clsJ[j - col0];
#pragma unroll
    for (int v = 0; v < 8; ++v) {
        const int i = ibase + v + 8 * hi;
        if (j > i) {                       // strict upper triangle
            float s    = c[v];
            float m    = floorf(s * INV);            // floor(s/STEP)
            float frac = (s - m * STEP) * INV;       // (s - f)/STEP
            int klow = (int)m + 75;                  // round((f+1)/STEP)
            klow = klow < 0 ? 0 : (klow > BINS - 1 ? BINS - 1 : klow);
            int kup = (klow + 1 > BINS - 1) ? BINS - 1 : klow + 1;
            float* h = (clsI[i - row0] == cj) ? hp : hn;
            atomicAdd(&h[klow], 1.0f - frac);        // ds_add_f32
            atomicAdd(&h[kup],  frac);
        }
    }
}

__global__ void __launch_bounds__(256)
gram_hist_kernel(const float* __restrict__ F, const int* __restrict__ cls,
                 float* __restrict__ histP, float* __restrict__ histN)
{
    __shared__ float hl[8][2][BINS_PAD];   // per-wave private pos/neg hists
    __shared__ int clsI[BM];
    __shared__ int clsJ[BN];

    const int row0 = blockIdx.y * BM;
    const int col0 = blockIdx.x * BN;
    // Entire tile strictly below the diagonal -> nothing to do. Uniform branch,
    // so surviving blocks keep EXEC all-ones for WMMA.
    if (col0 + BN - 1 <= row0) return;

    const int tid  = threadIdx.x;
    const int wave = tid >> 5;
    const int lane = tid & 31;

    for (int b = tid; b < 8 * 2 * BINS_PAD; b += 256) (&hl[0][0][0])[b] = 0.0f;
    if (tid < BM) clsI[tid] = cls[row0 + tid];
    if (tid < BN) clsJ[tid] = cls[col0 + tid];
    __syncthreads();

    // 8 waves in a 4x2 arrangement; each wave owns a 32x32 output sub-tile.
    const int wr = wave & 3;     // row group 0..3  -> rows  row0 + wr*32 ..
    const int wc = wave >> 2;    // col group 0..1  -> cols  col0 + wc*32 ..

    // A 16x4 f32 layout: lane L: row = L&15, K slots = {kq, kq+1}, kq = (L>>4)*2.
    // B 4x16 mirrors it, and B = F_cols^T, so both fragments load F rows.
    const int r16 = lane & 15;
    const int kq  = (lane >> 4) * 2;
    const float* Ar = F + (size_t)(row0 + wr * 32 + r16) * DD + kq;
    const float* Br = F + (size_t)(col0 + wc * 32 + r16) * DD + kq;

    v8f c00 = {}, c01 = {}, c10 = {}, c11 = {};
#pragma unroll 4
    for (int k0 = 0; k0 < DD; k0 += 4) {
        v2f a0 = *(const v2f*)(Ar + k0);             // rows wr*32 + 0..15
        v2f a1 = *(const v2f*)(Ar + 16 * DD + k0);   // rows wr*32 + 16..31
        v2f b0 = *(const v2f*)(Br + k0);             // cols wc*32 + 0..15
        v2f b1 = *(const v2f*)(Br + 16 * DD + k0);   // cols wc*32 + 16..31
        c00 = __builtin_amdgcn_wmma_f32_16x16x4_f32(false, a0, false, b0,
                                                    (short)0, c00, false, false);
        c01 = __builtin_amdgcn_wmma_f32_16x16x4_f32(false, a0, false, b1,
                                                    (short)0, c01, false, false);
        c10 = __builtin_amdgcn_wmma_f32_16x16x4_f32(false, a1, false, b0,
                                                    (short)0, c10, false, false);
        c11 = __builtin_amdgcn_wmma_f32_16x16x4_f32(false, a1, false, b1,
                                                    (short)0, c11, false, false);
    }

    float* hp = &hl[wave][0][0];
    float* hn = &hl[wave][1][0];
    const int ib = row0 + wr * 32;
    const int jb = col0 + wc * 32;
    bin_tile(c00, ib,      jb,      lane, clsI, clsJ, row0, col0, hp, hn);
    bin_tile(c01, ib,      jb + 16, lane, clsI, clsJ, row0, col0, hp, hn);
    bin_tile(c10, ib + 16, jb,      lane, clsI, clsJ, row0, col0, hp, hn);
    bin_tile(c11, ib + 16, jb + 16, lane, clsI, clsJ, row0, col0, hp, hn);
    __syncthreads();

    // Merge the 8 per-wave hists, one global atomic per touched bin.
    for (int b = tid; b < 2 * BINS_PAD; b += 256) {
        int sel = b / BINS_PAD, bin = b % BINS_PAD;
        if (bin < BINS) {
            float sum = 0.0f;
            for (int w = 0; w < 8; ++w) sum += hl[w][sel][bin];
            if (sum != 0.0f) atomicAdd((sel ? histN : histP) + bin, sum);
        }
    }
}

// ---------------------------------------------------------------- stage 4
__global__ void finalize_kernel(const float* __restrict__ histP,
                                const float* __restrict__ histN,
                                const unsigned* __restrict__ cnt,
                                float* __restrict__ out)
{
    if (threadIdx.x == 0) {
        double npos = 0.0;
        for (int c = 0; c < NCLS; ++c) {
            double m = (double)cnt[c];
            npos += 0.5 * m * (m - 1.0);
        }
        double total = 0.5 * (double)NN * (double)(NN - 1);
        float fpos = (float)(npos > 1.0 ? npos : 1.0);
        double nneg = total - npos;
        float fneg = (float)(nneg > 1.0 ? nneg : 1.0);
        float rpos = 1.0f / fpos;
        float rneg = 1.0f / fneg;
        float prefix = 0.0f, loss = 0.0f;
        for (int k = 0; k < BINS; ++k) {
            prefix += histP[k] * rpos;            // cdf of hist_pos
            loss   += (histN[k] * rneg) * prefix; // sum(hist_neg * cdf_pos)
        }
        out[0] = loss;
    }
}

// ---------------------------------------------------------------- launch
extern "C" void kernel_launch(void* const* d_in, const int* in_sizes, int n_in,
                              void* d_out, int out_size, void* d_ws, size_t ws_size,
                              hipStream_t stream) {
    const float* F   = (const float*)d_in[0];   // [8192, 256] f32, L2-normalized
    const int*   cls = (const int*)d_in[1];     // [8192] int32 (JAX default x64 off)
    float* out = (float*)d_out;                 // scalar f32

    float*    histP = (float*)d_ws;             // [BINS_PAD]
    float*    histN = histP + BINS_PAD;         // [BINS_PAD]
    unsigned* cnt   = (unsigned*)(histN + BINS_PAD);  // [NCLS]

    zero_ws_kernel<<<1, 512, 0, stream>>>((unsigned*)d_ws);
    class_count_kernel<<<NN / 256, 256, 0, stream>>>(cls, cnt);
    gram_hist_kernel<<<dim3(NN / BN, NN / BM), 256, 0, stream>>>(F, cls, histP, histN);
    finalize_kernel<<<1, 32, 0, stream>>>(histP, histN, cnt, out);
}